// CTCLoss_10093173145929
// MI455X (gfx1250) — compile-verified
//
#include <hip/hip_runtime.h>
#include <math.h>

// CTC loss forward: B=32, T=1024, C=512, S=256 (derived from in_sizes).
// Phase 1 (parallel, HBM-bound): per-(b,t) log-sum-exp + gather of extended-label
//   log-probs into lp[row][s] (log2 domain), staged through LDS with CDNA5
//   global_load_async_to_lds_b32 + s_wait_asynccnt.
// Phase 2 (serial scan): one wave32 per batch element. Lane l owns states
//   16l..16l+15 in registers (+ state 512 on lane 31). Neighbor exchange via two
//   __shfl_up per step; lp rows software-pipelined (load t+1 before update t).
//   Even states (blanks) use a compile-time 2-term logaddexp.

#define NEGV      (-1.0e9f)
#define LOG2E     1.4426950408889634f
#define INV_LOG2E 0.6931471805599453f
#define LP_STRIDE 516   // 513 rounded up; 516*4 bytes = 16B-aligned rows

#if __has_builtin(__builtin_amdgcn_global_load_async_to_lds_b32)
#define HAVE_ASYNC_LDS 1
#else
#define HAVE_ASYNC_LDS 0
#endif

__device__ __forceinline__ float fexp2f(float x) {
#if __has_builtin(__builtin_amdgcn_exp2f)
  return __builtin_amdgcn_exp2f(x);      // v_exp_f32 (base-2)
#else
  return exp2f(x);
#endif
}
__device__ __forceinline__ float flog2f(float x) {
#if __has_builtin(__builtin_amdgcn_logf)
  return __builtin_amdgcn_logf(x);       // v_log_f32 (base-2)
#else
  return log2f(x);
#endif
}
__device__ __forceinline__ void wait_async0() {
#if __has_builtin(__builtin_amdgcn_s_wait_asynccnt)
  __builtin_amdgcn_s_wait_asynccnt(0);
#else
  asm volatile("s_wait_asynccnt 0" ::: "memory");
#endif
}

__device__ __forceinline__ float lae2(float a, float b) {          // logaddexp, log2 domain
  float mx = fmaxf(a, b);
  return mx + flog2f(fexp2f(a - mx) + fexp2f(b - mx));
}
__device__ __forceinline__ float lae3(float a, float b, float c) {
  float mx = fmaxf(fmaxf(a, b), c);
  return mx + flog2f(fexp2f(a - mx) + fexp2f(b - mx) + fexp2f(c - mx));
}

// ---------------------------------------------------------------------------
// Phase 1: one block (256 threads) per (b,t) row.
// ---------------------------------------------------------------------------
__global__ __launch_bounds__(256)
void ctc_row_kernel(const float* __restrict__ pred,
                    const int*   __restrict__ phonemes,
                    float*       __restrict__ lse_out,   // may be null
                    float*       __restrict__ lp_out,    // may be null
                    int T, int C, int S, int L)
{
  const int row = blockIdx.x;          // b*T + t
  const int b   = row / T;
  const int tid = threadIdx.x;
  const int wid = tid >> 5;
  const int lid = tid & 31;

  __shared__ float xs[512];
  __shared__ float redbuf[8];
  __shared__ float sh_m, sh_lse;

  const float* prow = pred + (size_t)row * (size_t)C;

#if HAVE_ASYNC_LDS
  __builtin_amdgcn_global_load_async_to_lds_b32(
      (__attribute__((address_space(1))) int*)(prow + tid),
      (__attribute__((address_space(3))) int*)(&xs[tid]), 0, 0);
  __builtin_amdgcn_global_load_async_to_lds_b32(
      (__attribute__((address_space(1))) int*)(prow + tid + 256),
      (__attribute__((address_space(3))) int*)(&xs[tid + 256]), 0, 0);
  wait_async0();
  __syncthreads();
  float x0 = xs[tid];
  float x1 = xs[tid + 256];
#else
  float x0 = prow[tid];
  float x1 = prow[tid + 256];
  xs[tid]       = x0;
  xs[tid + 256] = x1;
  __syncthreads();
#endif

  // --- block max ---
  float m = fmaxf(x0, x1);
#pragma unroll
  for (int off = 16; off > 0; off >>= 1)
    m = fmaxf(m, __shfl_xor(m, off, 32));
  if (lid == 0) redbuf[wid] = m;
  __syncthreads();
  if (wid == 0) {
    float mm = (lid < 8) ? redbuf[lid] : -INFINITY;
#pragma unroll
    for (int off = 4; off > 0; off >>= 1)
      mm = fmaxf(mm, __shfl_xor(mm, off, 32));
    if (lid == 0) sh_m = mm;
  }
  __syncthreads();
  m = sh_m;

  // --- block sum of exp ---
  float ss = expf(x0 - m) + expf(x1 - m);
#pragma unroll
  for (int off = 16; off > 0; off >>= 1)
    ss += __shfl_xor(ss, off, 32);
  if (lid == 0) redbuf[wid] = ss;
  __syncthreads();
  if (tid == 0) {
    float tot = 0.0f;
#pragma unroll
    for (int w = 0; w < 8; ++w) tot += redbuf[w];
    float l = m + logf(tot);
    sh_lse = l;
    if (lse_out) lse_out[row] = l;
  }
  __syncthreads();

  if (lp_out) {
    const float lse = sh_lse;
    float* lprow = lp_out + (size_t)row * (size_t)LP_STRIDE;
    for (int s = tid; s < L; s += 256) {
      int lab = (s & 1) ? phonemes[b * S + (s >> 1)] : 0;
      lprow[s] = (xs[lab] - lse) * LOG2E;
    }
  }
}

// ---------------------------------------------------------------------------
// Phase 2 helpers.
// ---------------------------------------------------------------------------
struct Alpha {
  float a[16];
  float a512;   // state 512, meaningful on lane 31 only
};

// One CTC transition step (log2 domain). lpv[16] + lp512 are this step's log-probs.
__device__ __forceinline__ void ctc_step(Alpha& st, const float* lpv, float lp512,
                                         unsigned skipm, int lane)
{
  float last1 = __shfl_up(st.a[15], 1, 32);   // prev lane's s-1
  float last2 = __shfl_up(st.a[14], 1, 32);   // prev lane's s-2
  if (lane == 0) { last1 = NEGV; last2 = NEGV; }
  const float a15old = st.a[15];
  float na[16];
#pragma unroll
  for (int i = 0; i < 16; ++i) {
    float a0 = st.a[i];
    float a1 = (i >= 1) ? st.a[i - 1] : last1;
    if (i & 1) {                                   // odd state: label, may skip
      float a2s = (i >= 2) ? st.a[i - 2] : last1;  // i==1 -> prev lane's a[15]
      float a2  = ((skipm >> i) & 1u) ? a2s : NEGV;
      na[i] = lae3(a0, a1, a2) + lpv[i];
    } else {                                       // even state: blank, no skip
      float a2e = (i == 0) ? ((skipm & 1u) ? last2 : NEGV) : NEGV;
      // i==0 is even (s=16*lane even) => blank => skipm bit 0 never set; keep NEGV.
      (void)a2e;
      na[i] = lae2(a0, a1) + lpv[i];
    }
  }
  // state 512 (even/blank): a1 = old alpha[511] = lane31's old a[15]
  st.a512 = lae2(st.a512, a15old) + lp512;
#pragma unroll
  for (int i = 0; i < 16; ++i) st.a[i] = na[i];
}

// ---------------------------------------------------------------------------
// Phase 2: one wave32 per batch element.
// ---------------------------------------------------------------------------
__global__ __launch_bounds__(32)
void ctc_alpha_kernel(const float* __restrict__ pred,
                      const float* __restrict__ lse,   // gather mode w/ precomputed lse
                      const float* __restrict__ lp,    // null => gather mode
                      const int*   __restrict__ phonemes,
                      const int*   __restrict__ input_len,
                      const int*   __restrict__ target_len,
                      float*       __restrict__ partial,
                      int T, int C, int S, int L)
{
  const int b    = blockIdx.x;
  const int lane = threadIdx.x;

  __shared__ int   ph[256];
  __shared__ float af[520];

  for (int i = lane; i < S; i += 32) ph[i] = phonemes[b * S + i];
  __syncthreads();

  // Per-lane labels / skip mask for states s = lane*16 + i.
  int lab[16];
  unsigned skipm = 0;
#pragma unroll
  for (int i = 0; i < 16; ++i) {
    int s = lane * 16 + i;
    int l = 0;
    if (s < L && (s & 1)) {
      l = ph[s >> 1];
      if (s >= 3 && l != ph[(s >> 1) - 1]) skipm |= (1u << i);
    }
    lab[i] = l;
  }

  int Tb = input_len[b];
  if (Tb > T) Tb = T;
  if (Tb < 1) Tb = 1;

  Alpha st;
  st.a512 = NEGV;

  if (lp) {
    // ---------------- lp mode: pipelined b128 row loads ----------------
    const float* lprow0 = lp + (size_t)b * (size_t)T * (size_t)LP_STRIDE;
    float4 cur[4]; float cur512;
    {
      const float4* p4 = (const float4*)(lprow0 + lane * 16);
      cur[0] = p4[0]; cur[1] = p4[1]; cur[2] = p4[2]; cur[3] = p4[3];
      cur512 = lprow0[512];
    }
    for (int t = 0; t < Tb; ++t) {
      float4 nxt[4]; float nxt512 = 0.0f;
      const bool more = (t + 1 < Tb);
      if (more) {
        const float* lprow = lprow0 + (size_t)(t + 1) * (size_t)LP_STRIDE;
        const float4* p4 = (const float4*)(lprow + lane * 16);
        nxt[0] = p4[0]; nxt[1] = p4[1]; nxt[2] = p4[2]; nxt[3] = p4[3];
        nxt512 = lprow[512];
      }
      float lpv[16] = { cur[0].x, cur[0].y, cur[0].z, cur[0].w,
                        cur[1].x, cur[1].y, cur[1].z, cur[1].w,
                        cur[2].x, cur[2].y, cur[2].z, cur[2].w,
                        cur[3].x, cur[3].y, cur[3].z, cur[3].w };
#pragma unroll
      for (int i = 0; i < 16; ++i) {       // guard padded states (only if L<513)
        int s = lane * 16 + i;
        if (s >= L) lpv[i] = NEGV;
      }
      if (t == 0) {
#pragma unroll
        for (int i = 0; i < 16; ++i) {
          int s = lane * 16 + i;
          st.a[i] = (s < 2) ? lpv[i] : NEGV;
        }
      } else {
        ctc_step(st, lpv, cur512, skipm, lane);
      }
      if (more) {
        cur[0] = nxt[0]; cur[1] = nxt[1]; cur[2] = nxt[2]; cur[3] = nxt[3];
        cur512 = nxt512;
      }
    }
  } else {
    // ---------------- gather modes ----------------
    for (int t = 0; t < Tb; ++t) {
      const size_t row = (size_t)b * (size_t)T + (size_t)t;
      const float* prow = pred + row * (size_t)C;
      float ls;
      if (lse) {
        ls = lse[row];
      } else {
        float v[16];
        float mx = -INFINITY;
#pragma unroll
        for (int k = 0; k < 16; ++k) { v[k] = prow[lane + 32 * k]; mx = fmaxf(mx, v[k]); }
#pragma unroll
        for (int off = 16; off > 0; off >>= 1) mx = fmaxf(mx, __shfl_xor(mx, off, 32));
        float sm = 0.0f;
#pragma unroll
        for (int k = 0; k < 16; ++k) sm += expf(v[k] - mx);
#pragma unroll
        for (int off = 16; off > 0; off >>= 1) sm += __shfl_xor(sm, off, 32);
        ls = mx + logf(sm);
      }
      float lpv[16];
#pragma unroll
      for (int i = 0; i < 16; ++i) {
        int s = lane * 16 + i;
        lpv[i] = (s < L) ? (prow[lab[i]] - ls) * LOG2E : NEGV;
      }
      float lp512 = (prow[0] - ls) * LOG2E;   // state 512 is a blank
      if (t == 0) {
#pragma unroll
        for (int i = 0; i < 16; ++i) {
          int s = lane * 16 + i;
          st.a[i] = (s < 2) ? lpv[i] : NEGV;
        }
      } else {
        ctc_step(st, lpv, lp512, skipm, lane);
      }
    }
  }

#pragma unroll
  for (int i = 0; i < 16; ++i) {
    int s = lane * 16 + i;
    if (s < L) af[s] = st.a[i];
  }
  if (lane == 31 && L > 512) af[512] = st.a512;
  __syncthreads();

  if (lane == 0) {
    int Sb = target_len[b];
    int le = 2 * Sb;                       // <= 512
    float A  = af[le];
    float Bv = af[le - 1];
    float ll = lae2(A, Bv) * INV_LOG2E;    // back to natural log
    partial[b] = -ll / (float)Sb;
  }
}

// ---------------------------------------------------------------------------
// Deterministic ordered mean of per-batch partials.
// ---------------------------------------------------------------------------
__global__ void ctc_finalize_kernel(const float* __restrict__ partial,
                                    float* __restrict__ out, int B)
{
  if (threadIdx.x == 0 && blockIdx.x == 0) {
    float s = 0.0f;
    for (int b = 0; b < B; ++b) s += partial[b];
    out[0] = s / (float)B;
  }
}

// ---------------------------------------------------------------------------
extern "C" void kernel_launch(void* const* d_in, const int* in_sizes, int n_in,
                              void* d_out, int out_size, void* d_ws, size_t ws_size,
                              hipStream_t stream)
{
  (void)n_in; (void)out_size;
  const float* pred     = (const float*)d_in[0];
  const int*   phonemes = (const int*)  d_in[1];
  const int*   text_len = (const int*)  d_in[2];  // input lengths (time)
  const int*   phon_len = (const int*)  d_in[3];  // target lengths

  const int B = in_sizes[2];
  const int S = in_sizes[1] / B;
  const int C = 512;
  const int T = in_sizes[0] / (B * C);
  const int L = 2 * S + 1;

  float* partial = (float*)d_ws;                       // B floats (pad to 256B)
  char*  base    = (char*)d_ws + 256;
  const size_t lse_bytes = (size_t)B * T * sizeof(float);
  const size_t lp_bytes  = (size_t)B * T * LP_STRIDE * sizeof(float);

  const bool use_lp  = (ws_size >= 256 + lp_bytes) && (S <= 256) && (L <= 513);
  const bool use_lse = !use_lp && (ws_size >= 256 + lse_bytes);

  const int rows = B * T;

  if (use_lp) {
    float* lp = (float*)base;
    ctc_row_kernel<<<rows, 256, 0, stream>>>(pred, phonemes, nullptr, lp, T, C, S, L);
    ctc_alpha_kernel<<<B, 32, 0, stream>>>(pred, nullptr, lp, phonemes,
                                           text_len, phon_len, partial, T, C, S, L);
  } else if (use_lse) {
    float* lse = (float*)base;
    ctc_row_kernel<<<rows, 256, 0, stream>>>(pred, phonemes, lse, nullptr, T, C, S, L);
    ctc_alpha_kernel<<<B, 32, 0, stream>>>(pred, lse, nullptr, phonemes,
                                           text_len, phon_len, partial, T, C, S, L);
  } else {
    ctc_alpha_kernel<<<B, 32, 0, stream>>>(pred, nullptr, nullptr, phonemes,
                                           text_len, phon_len, partial, T, C, S, L);
  }

  ctc_finalize_kernel<<<1, 32, 0, stream>>>(partial, (float*)d_out, B);
}